// TranslationLayer_63350767616318
// MI455X (gfx1250) — compile-verified
//
#include <hip/hip_runtime.h>
#include <hip/hip_bf16.h>

// Problem constants (match the JAX reference)
#define BATCH  128
#define HEIGHT 512
#define WIDTH  512   // C == 1, folded away
#define RPB    8     // rows per block (async pipeline depth)

typedef float v4f __attribute__((ext_vector_type(4)));

// One block = 8 output rows of one sample. 128 threads * 4 floats = 512-wide row.
// Pipeline: issue 8 async global->LDS b128 row transfers back-to-back
// (4 KB in flight per wave on ASYNCcnt), drain once, barrier once, then
// resolve the horizontal shift in LDS and stream out with nontemporal b128.
__global__ __launch_bounds__(128) void TranslationLayer_63350767616318_kernel(
    const float* __restrict__ in,
    const int*   __restrict__ dx,
    const int*   __restrict__ dy,
    float*       __restrict__ out)
{
    __shared__ float rows[RPB][WIDTH];   // 16 KB of the 320 KB/WGP LDS

    const int i0 = blockIdx.x * RPB;     // first output row of this block
    const int b  = blockIdx.y;           // sample
    const int t  = threadIdx.x;          // 0..127
    const int j0 = t << 2;               // first of 4 output columns

    // Uniform per block -> scalar (SMEM) loads
    const int dxb = dx[b];
    const int dyb = dy[b];

    const float* src_base = in + (size_t)b * HEIGHT * WIDTH;

    // Stage up to 8 source rows into LDS. Out-of-bounds rows skip their issue
    // (block-uniform scalar branch; their buffer is garbage but never read).
    // Draining with asynccnt==0 makes the per-row issue count irrelevant.
#pragma unroll
    for (int r = 0; r < RPB; ++r) {
        const int src_i = i0 + r + dyb;
        if (src_i >= 0 && src_i < HEIGHT) {
            unsigned     lds = (unsigned)(uintptr_t)(&rows[r][j0]);
            const float* g   = src_base + (size_t)src_i * WIDTH + j0;
            asm volatile("global_load_async_to_lds_b128 %0, %1, off"
                         :
                         : "v"(lds), "v"(g)
                         : "memory");
        }
    }

    // One drain + one barrier per 8 rows: each wave drains its own ASYNCcnt,
    // the barrier makes every wave's quarter-row visible block-wide.
    asm volatile("s_wait_asynccnt 0x0" ::: "memory");
    __syncthreads();

    float* out_base = out + ((size_t)b * HEIGHT + (size_t)i0) * WIDTH + j0;

    // Shift resolved in LDS: DWORD reads have no alignment penalty and stay
    // bank-conflict-free under a uniform lane shift.
#pragma unroll
    for (int r = 0; r < RPB; ++r) {
        const int src_i = i0 + r + dyb;
        v4f v = {0.0f, 0.0f, 0.0f, 0.0f};
        if (src_i >= 0 && src_i < HEIGHT) {
            const int sj = j0 - dxb;
            v.x = ((unsigned)(sj    ) < WIDTH) ? rows[r][sj    ] : 0.0f;
            v.y = ((unsigned)(sj + 1) < WIDTH) ? rows[r][sj + 1] : 0.0f;
            v.z = ((unsigned)(sj + 2) < WIDTH) ? rows[r][sj + 2] : 0.0f;
            v.w = ((unsigned)(sj + 3) < WIDTH) ? rows[r][sj + 3] : 0.0f;
        }
        // Streaming 128-bit store; never re-read -> keep near caches clean.
        __builtin_nontemporal_store(v, (v4f*)(out_base + (size_t)r * WIDTH));
    }
}

extern "C" void kernel_launch(void* const* d_in, const int* in_sizes, int n_in,
                              void* d_out, int out_size, void* d_ws, size_t ws_size,
                              hipStream_t stream)
{
    const float* inputs = (const float*)d_in[0];
    const int*   dx     = (const int*)d_in[1];
    const int*   dy     = (const int*)d_in[2];
    float*       out    = (float*)d_out;

    dim3 grid(HEIGHT / RPB, BATCH);  // (64, 128) blocks, 8 rows each
    dim3 block(128);                 // 4 wave32s, one 512-float row slice each
    TranslationLayer_63350767616318_kernel<<<grid, block, 0, stream>>>(
        inputs, dx, dy, out);
}